// CrowdCountingLoss_36893769073045
// MI455X (gfx1250) — compile-verified
//
#include <hip/hip_runtime.h>
#include <hip/hip_bf16.h>
#include <math.h>

// ---------------------------------------------------------------------------
// CrowdCountingLoss for MI455X (gfx1250, wave32).
//   out = BETA*mean((pred-gt_blur)^2) + BETA*|pc-gc| + BETA*ALPHA*sinkhorn
// Sinkhorn on the fixed 64x64 grid, exploiting the Kronecker-separable
// squared-Euclidean cost: every LSE over 4096 points factors into two LSEs
// over 64 points (32x work reduction). All state lives in LDS (one WGP),
// stages separated by workgroup barriers. Density MSE partial sums run on
// the matrix pipe via v_wmma_f32_16x16x4_f32 with a ones B-matrix.
// ---------------------------------------------------------------------------

typedef __attribute__((ext_vector_type(2))) float v2f;
typedef __attribute__((ext_vector_type(8))) float v8f;

#define HW        64
#define NPIX      4096
#define LDP       65            // padded LDS row pitch -> conflict-free columns
#define NTHREADS  1024          // 32 waves (wave32) on one WGP
#define SINK_IT   20
#define INV_EPS   100.0f        // 1/eps, eps = 0.01
#define ALPHA_F   3.6e-07f
#define BETA_F    8e-05f
#define NEG_BIG   -3.0e38f
#define REL_SKIP  -25.0f        // exp(x), x<=-25 cannot move an fp32 sum >= 1
#define ABS_SKIP  -80.0f        // exp underflow guard for weighted sums

__global__ __launch_bounds__(NTHREADS)
void crowd_loss_kernel(const float* __restrict__ pred,
                       const float* __restrict__ gt,      // unused numerically
                       const float* __restrict__ gt_blur,
                       const float* __restrict__ pcnt,
                       const float* __restrict__ gcnt,
                       float* __restrict__ out)
{
    __shared__ float U[HW * LDP];   // u / eps
    __shared__ float V[HW * LDP];   // v / eps
    __shared__ float T[HW * LDP];   // factorized-LSE scratch
    __shared__ float red[32];
    __shared__ float acc[2];        // [0]=density mean, [1]=count loss

    const int tid  = (int)threadIdx.x;
    const int lane = tid & 31;
    const int wid  = tid >> 5;
    (void)gt;

    // ---------------- Phase 0: density MSE partial sums on the matrix pipe --
    // Each wave reduces 128 squared diffs with two chained WMMAs:
    // A = 16x4 tile of d^2, B = ones(4x16)  =>  D rows hold row-sums (x16).
    v8f cacc = {0.f, 0.f, 0.f, 0.f, 0.f, 0.f, 0.f, 0.f};
    v2f onesb; onesb.x = 1.0f; onesb.y = 1.0f;
#pragma unroll
    for (int t = 0; t < 2; ++t) {
        int base = wid * 128 + t * 64 + lane * 2;
        float d0 = pred[base]     - gt_blur[base];
        float d1 = pred[base + 1] - gt_blur[base + 1];
        v2f a; a.x = d0 * d0; a.y = d1 * d1;
        cacc = __builtin_amdgcn_wmma_f32_16x16x4_f32(
                   false, a, false, onesb, (short)0, cacc, false, false);
    }
    float wsum = cacc[0] + cacc[1] + cacc[2] + cacc[3]
               + cacc[4] + cacc[5] + cacc[6] + cacc[7];
#pragma unroll
    for (int off = 16; off > 0; off >>= 1)
        wsum += __shfl_xor(wsum, off, 32);
    if (lane == 0) red[wid] = wsum * (1.0f / 16.0f);   // undo 16x replication
    __syncthreads();
    if (tid == 0) {
        float s = 0.f;
        for (int w = 0; w < 32; ++w) s += red[w];
        acc[0] = s * (1.0f / (float)NPIX);
        acc[1] = fabsf(pcnt[0] - gcnt[0]);
    }

    // ---------------- init scaled potentials -------------------------------
    for (int i = tid; i < HW * LDP; i += NTHREADS) { U[i] = 0.f; V[i] = 0.f; }
    __syncthreads();

    // ---------------- Sinkhorn iterations (factorized log-domain) ----------
    for (int it = 0; it < SINK_IT; ++it) {
        // Stage A: T[jr][ic] = LSE_jc( V[jr][jc] - 100*(ic-jc)^2 )
        for (int o = tid; o < NPIX; o += NTHREADS) {
            int jr = o >> 6, ic = o & 63;
            const float* vrow = &V[jr * LDP];
            float m = NEG_BIG;
#pragma unroll 8
            for (int jc = 0; jc < HW; ++jc) {
                int di = ic - jc;
                float x = __builtin_fmaf(-INV_EPS, (float)(di * di), vrow[jc]);
                m = fmaxf(m, x);
            }
            float s = 0.f;
#pragma unroll 8
            for (int jc = 0; jc < HW; ++jc) {
                int di = ic - jc;
                float e = __builtin_fmaf(-INV_EPS, (float)(di * di), vrow[jc]) - m;
                if (e > REL_SKIP) s += __expf(e);
            }
            T[jr * LDP + ic] = m + __logf(s);
        }
        __syncthreads();

        // Stage B: U[ir][ic] = -LSE_jr( T[jr][ic] - 100*(ir-jr)^2 )
        for (int o = tid; o < NPIX; o += NTHREADS) {
            int ir = o >> 6, ic = o & 63;
            float m = NEG_BIG;
#pragma unroll 8
            for (int jr = 0; jr < HW; ++jr) {
                int di = ir - jr;
                float x = __builtin_fmaf(-INV_EPS, (float)(di * di), T[jr * LDP + ic]);
                m = fmaxf(m, x);
            }
            float s = 0.f;
#pragma unroll 8
            for (int jr = 0; jr < HW; ++jr) {
                int di = ir - jr;
                float e = __builtin_fmaf(-INV_EPS, (float)(di * di), T[jr * LDP + ic]) - m;
                if (e > REL_SKIP) s += __expf(e);
            }
            U[ir * LDP + ic] = -(m + __logf(s));
        }
        __syncthreads();

        // Stage C: T[ir][jc] = LSE_ic( U[ir][ic] - 100*(ic-jc)^2 )
        for (int o = tid; o < NPIX; o += NTHREADS) {
            int ir = o >> 6, jc = o & 63;
            const float* urow = &U[ir * LDP];
            float m = NEG_BIG;
#pragma unroll 8
            for (int ic = 0; ic < HW; ++ic) {
                int di = ic - jc;
                float x = __builtin_fmaf(-INV_EPS, (float)(di * di), urow[ic]);
                m = fmaxf(m, x);
            }
            float s = 0.f;
#pragma unroll 8
            for (int ic = 0; ic < HW; ++ic) {
                int di = ic - jc;
                float e = __builtin_fmaf(-INV_EPS, (float)(di * di), urow[ic]) - m;
                if (e > REL_SKIP) s += __expf(e);
            }
            T[ir * LDP + jc] = m + __logf(s);
        }
        __syncthreads();

        // Stage D: V[jr][jc] = -LSE_ir( T[ir][jc] - 100*(ir-jr)^2 )
        for (int o = tid; o < NPIX; o += NTHREADS) {
            int jr = o >> 6, jc = o & 63;
            float m = NEG_BIG;
#pragma unroll 8
            for (int ir = 0; ir < HW; ++ir) {
                int di = ir - jr;
                float x = __builtin_fmaf(-INV_EPS, (float)(di * di), T[ir * LDP + jc]);
                m = fmaxf(m, x);
            }
            float s = 0.f;
#pragma unroll 8
            for (int ir = 0; ir < HW; ++ir) {
                int di = ir - jr;
                float e = __builtin_fmaf(-INV_EPS, (float)(di * di), T[ir * LDP + jc]) - m;
                if (e > REL_SKIP) s += __expf(e);
            }
            V[jr * LDP + jc] = -(m + __logf(s));
        }
        __syncthreads();
    }

    // ---------------- Final transport cost  sum(P*C) = S1 + S2 -------------
    float sink = 0.f;

    // Stage E: T[jr][ic] = LSE_jc( V[jr][jc] - 100*(ic-jc)^2 )
    for (int o = tid; o < NPIX; o += NTHREADS) {
        int jr = o >> 6, ic = o & 63;
        const float* vrow = &V[jr * LDP];
        float m = NEG_BIG;
#pragma unroll 8
        for (int jc = 0; jc < HW; ++jc) {
            int di = ic - jc;
            float x = __builtin_fmaf(-INV_EPS, (float)(di * di), vrow[jc]);
            m = fmaxf(m, x);
        }
        float s = 0.f;
#pragma unroll 8
        for (int jc = 0; jc < HW; ++jc) {
            int di = ic - jc;
            float e = __builtin_fmaf(-INV_EPS, (float)(di * di), vrow[jc]) - m;
            if (e > REL_SKIP) s += __expf(e);
        }
        T[jr * LDP + ic] = m + __logf(s);
    }
    __syncthreads();

    // Stage F: S1 = sum_{ir,jr} dr2 * exp( LSE_ic(U[ir][ic]+T[jr][ic]) - 100*dr2 )
    for (int o = tid; o < NPIX; o += NTHREADS) {
        int ir = o >> 6, jr = o & 63;
        const float* urow = &U[ir * LDP];
        float m = NEG_BIG;
#pragma unroll 8
        for (int ic = 0; ic < HW; ++ic)
            m = fmaxf(m, urow[ic] + T[jr * LDP + ic]);
        float s = 0.f;
#pragma unroll 8
        for (int ic = 0; ic < HW; ++ic) {
            float e = urow[ic] + T[jr * LDP + ic] - m;
            if (e > REL_SKIP) s += __expf(e);
        }
        float logA = m + __logf(s);
        int di = ir - jr;
        float dr2 = (float)(di * di);
        float arg = __builtin_fmaf(-INV_EPS, dr2, logA);
        if (arg > ABS_SKIP) sink += dr2 * __expf(arg);
    }
    __syncthreads();

    // Stage G: T[ir][jc] = LSE_jr( V[jr][jc] - 100*(ir-jr)^2 )
    for (int o = tid; o < NPIX; o += NTHREADS) {
        int ir = o >> 6, jc = o & 63;
        float m = NEG_BIG;
#pragma unroll 8
        for (int jr = 0; jr < HW; ++jr) {
            int di = ir - jr;
            float x = __builtin_fmaf(-INV_EPS, (float)(di * di), V[jr * LDP + jc]);
            m = fmaxf(m, x);
        }
        float s = 0.f;
#pragma unroll 8
        for (int jr = 0; jr < HW; ++jr) {
            int di = ir - jr;
            float e = __builtin_fmaf(-INV_EPS, (float)(di * di), V[jr * LDP + jc]) - m;
            if (e > REL_SKIP) s += __expf(e);
        }
        T[ir * LDP + jc] = m + __logf(s);
    }
    __syncthreads();

    // Stage H: S2 = sum_{ic,jc} dc2 * exp( LSE_ir(U[ir][ic]+T[ir][jc]) - 100*dc2 )
    for (int o = tid; o < NPIX; o += NTHREADS) {
        int ic = o >> 6, jc = o & 63;
        float m = NEG_BIG;
#pragma unroll 8
        for (int ir = 0; ir < HW; ++ir)
            m = fmaxf(m, U[ir * LDP + ic] + T[ir * LDP + jc]);
        float s = 0.f;
#pragma unroll 8
        for (int ir = 0; ir < HW; ++ir) {
            float e = U[ir * LDP + ic] + T[ir * LDP + jc] - m;
            if (e > REL_SKIP) s += __expf(e);
        }
        float logB = m + __logf(s);
        int di = ic - jc;
        float dc2 = (float)(di * di);
        float arg = __builtin_fmaf(-INV_EPS, dc2, logB);
        if (arg > ABS_SKIP) sink += dc2 * __expf(arg);
    }
    __syncthreads();

    // ---------------- deterministic final reduction ------------------------
#pragma unroll
    for (int off = 16; off > 0; off >>= 1)
        sink += __shfl_xor(sink, off, 32);
    if (lane == 0) red[wid] = sink;
    __syncthreads();
    if (tid == 0) {
        float S = 0.f;
        for (int w = 0; w < 32; ++w) S += red[w];
        out[0] = BETA_F * acc[0] + BETA_F * acc[1] + (BETA_F * ALPHA_F) * S;
    }
}

extern "C" void kernel_launch(void* const* d_in, const int* in_sizes, int n_in,
                              void* d_out, int out_size, void* d_ws, size_t ws_size,
                              hipStream_t stream) {
    (void)in_sizes; (void)n_in; (void)out_size; (void)d_ws; (void)ws_size;
    const float* pred    = (const float*)d_in[0];
    const float* gt      = (const float*)d_in[1];
    const float* gt_blur = (const float*)d_in[2];
    const float* pcnt    = (const float*)d_in[3];
    const float* gcnt    = (const float*)d_in[4];
    float* out = (float*)d_out;
    crowd_loss_kernel<<<1, NTHREADS, 0, stream>>>(pred, gt, gt_blur, pcnt, gcnt, out);
}